// GEVCanonicalLoss_75076028334877
// MI455X (gfx1250) — compile-verified
//
#include <hip/hip_runtime.h>
#include <math.h>

typedef __attribute__((ext_vector_type(2))) float v2f;
typedef __attribute__((ext_vector_type(8))) float v8f;
typedef __attribute__((ext_vector_type(4))) float f32x4;
typedef __attribute__((ext_vector_type(4))) int   i32x4;

#define GEV_EPS   1e-6f
#define SQRT_PI_F 1.7724538509055160f

// erfc(s) ~= t*P(t) * exp(-s*s),  t = 1/(1+0.3275911*s)   (A&S 7.1.26, |err|<1.5e-7)
// Returns the polynomial factor only; caller multiplies by exp(-s*s).
__device__ __forceinline__ float erfc_poly(float s) {
  float t = __builtin_amdgcn_rcpf(fmaf(0.3275911f, s, 1.0f));
  float p = 1.061405429f;
  p = fmaf(p, t, -1.453152027f);
  p = fmaf(p, t,  1.421413741f);
  p = fmaf(p, t, -0.284496736f);
  p = fmaf(p, t,  0.254829592f);
  return p * t;
}

// l-term of the loss for one element (xi = 0.5 hardcoded):
//   base = max(1 + 0.5 x, eps); p = clamp(exp(-base^-2), eps, 1-eps); u = -log p
//   gammainc_neg(u) = 2 u^{-1/2} e^{-u} - 2 sqrt(pi) erfc(sqrt(u));  e^{-u} == p
__device__ __forceinline__ float gev_l(float x, float t) {
  float base = fmaxf(fmaf(0.5f, x, 1.0f), GEV_EPS);
  float ib   = __builtin_amdgcn_rcpf(base);
  float v    = ib * ib;                       // base^(-2)
  float p    = __expf(-v);
  p = fminf(fmaxf(p, GEV_EPS), 1.0f - GEV_EPS);
  float u    = -__logf(p);
  float eu   = p;                             // exp(-u) with u = -log(p)
  float rsu  = __builtin_amdgcn_rsqf(u);
  float su   = __builtin_amdgcn_sqrtf(u);
  float gn   = eu * fmaf(-2.0f * SQRT_PI_F, erfc_poly(su), 2.0f * rsu);
  return gn - t * fmaxf(x, -2.0f);            // clip(x, -1/xi, inf) = max(x, -2)
}

// Entropy term depends only on t in {0,1}: two uniform constants.
//   ent(t) = 2*(sqrt(pi)*erfc(sqrt(-log tc)) - t),  tc = clamp(t, eps, 1-eps)
__device__ __forceinline__ void ent_consts(float& ent0, float& ent1) {
  float u0 = -__logf(GEV_EPS);
  ent0 = 2.0f * SQRT_PI_F * erfc_poly(__builtin_amdgcn_sqrtf(u0)) * __expf(-u0);
  float u1 = -__logf(1.0f - GEV_EPS);
  ent1 = fmaf(2.0f * SQRT_PI_F * __expf(-u1),
              erfc_poly(__builtin_amdgcn_sqrtf(u1)), -2.0f);
}

// Reduce the 32 lane partials of a wave; each lane contributes (a0 + a1).
// Uses V_WMMA_F32_16X16X4_F32 with an all-ones B: row m of A is
// [a0(m), a1(m), a0(m+16), a1(m+16)] per the CDNA5 16x4 F32 A layout, so
// D[m][*] = r(m) = pairwise lane sum; summing D's 8 VGPRs gives rows 0-7 in
// lanes 0-15 and rows 8-15 in lanes 16-31; one xor-16 add completes it.
__device__ __forceinline__ float wave_reduce(float a0, float a1) {
#if __has_builtin(__builtin_amdgcn_wmma_f32_16x16x4_f32)
  v2f A; A.x = a0; A.y = a1;
  v2f B; B.x = 1.0f; B.y = 1.0f;
  v8f C = {0.0f, 0.0f, 0.0f, 0.0f, 0.0f, 0.0f, 0.0f, 0.0f};
  v8f D = __builtin_amdgcn_wmma_f32_16x16x4_f32(
      /*neg_a=*/false, A, /*neg_b=*/false, B,
      /*c_mod=*/(short)0, C, /*reuse_a=*/false, /*reuse_b=*/false);
  float s = ((D[0] + D[1]) + (D[2] + D[3])) + ((D[4] + D[5]) + (D[6] + D[7]));
  s += __shfl_xor(s, 16, 32);
  return s;
#else
  float s = a0 + a1;
  for (int m = 1; m < 32; m <<= 1) s += __shfl_xor(s, m, 32);
  return s;
#endif
}

__global__ __launch_bounds__(256) void gev_partial_kernel(
    const f32x4* __restrict__ inp, const i32x4* __restrict__ tgt,
    float* __restrict__ part, int n4) {
  float ent0, ent1;
  ent_consts(ent0, ent1);
  float acc0 = 0.0f, acc1 = 0.0f, acc2 = 0.0f, acc3 = 0.0f;
  const int stride = (int)(gridDim.x * blockDim.x);
  for (int i = (int)(blockIdx.x * blockDim.x + threadIdx.x); i < n4; i += stride) {
    f32x4 x  = __builtin_nontemporal_load(&inp[i]);   // TH=NT: stream past L2
    i32x4 tv = __builtin_nontemporal_load(&tgt[i]);
    acc0 += gev_l(x[0], (float)tv[0]) + (tv[0] ? ent1 : ent0);
    acc1 += gev_l(x[1], (float)tv[1]) + (tv[1] ? ent1 : ent0);
    acc2 += gev_l(x[2], (float)tv[2]) + (tv[2] ? ent1 : ent0);
    acc3 += gev_l(x[3], (float)tv[3]) + (tv[3] ? ent1 : ent0);
  }
  // All 256 threads reach here (reconverged) -> EXEC all ones for WMMA.
  float s = wave_reduce(acc0 + acc1, acc2 + acc3);

  __shared__ float red[32];
  const int lane = (int)(threadIdx.x & 31);
  const int wid  = (int)(threadIdx.x >> 5);
  if (lane == 0) red[wid] = s;
  __syncthreads();
  if (threadIdx.x == 0) {
    float bs = 0.0f;
    const int nw = (int)(blockDim.x >> 5);
    for (int w = 0; w < nw; ++w) bs += red[w];
    part[blockIdx.x] = bs;
  }
}

__global__ __launch_bounds__(256) void gev_final_kernel(
    const float* __restrict__ part, int nparts,
    const float* __restrict__ inp, const int* __restrict__ tgt,
    int n, int tail_start, float* __restrict__ out) {
  __shared__ float red[256];
  float s = 0.0f;
  for (int i = (int)threadIdx.x; i < nparts; i += (int)blockDim.x) s += part[i];
  if (threadIdx.x == 0) {  // scalar tail (n not divisible by 4)
    float ent0, ent1;
    ent_consts(ent0, ent1);
    for (int i = tail_start; i < n; ++i) {
      int t = tgt[i];
      s += gev_l(inp[i], (float)t) + (t ? ent1 : ent0);
    }
  }
  red[threadIdx.x] = s;
  __syncthreads();
  for (int off = 128; off > 0; off >>= 1) {
    if ((int)threadIdx.x < off) red[threadIdx.x] += red[threadIdx.x + off];
    __syncthreads();
  }
  if (threadIdx.x == 0) out[0] = red[0] / (float)n;
}

extern "C" void kernel_launch(void* const* d_in, const int* in_sizes, int n_in,
                              void* d_out, int out_size, void* d_ws, size_t ws_size,
                              hipStream_t stream) {
  const float* inp = (const float*)d_in[0];
  const int*   tgt = (const int*)d_in[1];
  float* out = (float*)d_out;

  const int n    = in_sizes[0];
  const int n4   = n >> 2;
  const int tail = n4 << 2;

  int nblocks = 2048;  // 2048 blocks x 8 wave32 -> plenty of HBM-latency cover
  size_t need = (size_t)nblocks * sizeof(float);
  if (ws_size < need) nblocks = (int)(ws_size / sizeof(float));
  int maxb = (n4 + 255) / 256;
  if (maxb < 1) maxb = 1;
  if (nblocks > maxb) nblocks = maxb;
  if (nblocks < 1) nblocks = 1;

  float* part = (float*)d_ws;
  gev_partial_kernel<<<dim3(nblocks), dim3(256), 0, stream>>>(
      (const f32x4*)inp, (const i32x4*)tgt, part, n4);
  gev_final_kernel<<<dim3(1), dim3(256), 0, stream>>>(
      part, nblocks, inp, tgt, n, tail, out);
}